// PatchSampleMLP_15839839387757
// MI455X (gfx1250) — compile-verified
//
#include <hip/hip_runtime.h>
#include <math.h>

// PatchSampleMLP for MI455X (gfx1250): gather + 2-layer MLP + L2 norm,
// fused per 32-row stripe, f32 WMMA (v_wmma_f32_16x16x4_f32).
// Gather is double-buffered: async global->LDS (ASYNCcnt) when available,
// otherwise register-staged software pipelining.

typedef float v2f __attribute__((ext_vector_type(2)));
typedef float v8f __attribute__((ext_vector_type(8)));

#if __has_builtin(__builtin_amdgcn_global_load_async_to_lds_b32) && \
    __has_builtin(__builtin_amdgcn_s_wait_asynccnt)
#define USE_ASYNC_GATHER 1
typedef __attribute__((address_space(1))) int g_int;   // global AS ("__device__ int*")
typedef __attribute__((address_space(3))) int l_int;   // LDS AS
#else
#define USE_ASYNC_GATHER 0
#endif

namespace {
constexpr int NC  = 256;        // hidden/output channels
constexpr int MT  = 32;         // rows per block
constexpr int KC  = 128;        // K chunk staged in LDS
constexpr int SAP = KC + 1;     // padded LDS stride (bank-conflict-free)
constexpr int SHP = NC + 1;
constexpr int EPT = KC / 8;     // gathered elements per thread per chunk (16)

struct Params {
    const float* feat[3];
    const int*   pid[3];
    const float* w1[3];
    const float* b1[3];
    const float* w2[3];
    const float* b2[3];
    float*       out;
    int          C[3];
    int          HW[3];
};
} // namespace

__global__ __launch_bounds__(256)
void psm_fused_kernel(Params p)
{
    const int s = blockIdx.y;                  // scale 0..2
    const float* __restrict__ feat = p.feat[s];
    const int*   __restrict__ pid  = p.pid[s];
    const float* __restrict__ w1   = p.w1[s];
    const float* __restrict__ b1   = p.b1[s];
    const float* __restrict__ w2   = p.w2[s];
    const float* __restrict__ b2   = p.b2[s];
    const int C  = p.C[s];
    const int HW = p.HW[s];
    float* __restrict__ out = p.out + (size_t)s * 4096 * NC;

    __shared__ float sA[2][MT * SAP];          // double-buffered gather chunk
    __shared__ float sH[MT * SHP];             // hidden (then final x)
    __shared__ unsigned long long sBase[MT];   // per-row element offset into feat
    __shared__ float sRed[MT * 8];
    __shared__ float sRnorm[MT];

    const int tid  = threadIdx.x;
    const int lane = tid & 31;
    const int w    = tid >> 5;                 // wave 0..7
    const int h    = lane >> 4;                // lane half (0/1)
    const int l16  = lane & 15;
    const int rt   = w & 1;                    // row tile (16 rows each)
    const int cg   = w >> 1;                   // col group (64 cols each)
    const int nbase = (cg << 6) + l16;
    const int m0   = blockIdx.x * MT;

    // Per-row gather bases: row m -> (b = m/256, patch = m%256)
    if (tid < MT) {
        const int mg = m0 + tid;
        sBase[tid] = (unsigned long long)(mg >> 8) * (unsigned long long)C
                         * (unsigned long long)HW
                   + (unsigned long long)(unsigned)pid[mg & 255];
    }

    const v8f vzero = {0.f, 0.f, 0.f, 0.f, 0.f, 0.f, 0.f, 0.f};
    v8f acc[4];
#pragma unroll
    for (int t = 0; t < 4; ++t) acc[t] = vzero;

    __syncthreads();

    const int r    = tid & 31;                 // gather row this thread serves
    const int cofs = tid >> 5;                 // gather col phase 0..7
    const float* __restrict__ rowp = feat + sBase[r];
    const int arow0 = ((rt << 4) + l16) * SAP + (h << 1);   // A-frag base (k added)

    // ---------------- GEMM1: x = feat_gather @ w1, K-chunked, pipelined -----
#if USE_ASYNC_GATHER
    int bb = 0;
    // Prologue: kick off chunk 0 into buffer 0 (scattered lane gather -> LDS).
#pragma unroll
    for (int e = 0; e < EPT; ++e) {
        const int c = (e << 3) + cofs;
        __builtin_amdgcn_global_load_async_to_lds_b32(
            (g_int*)(rowp + (size_t)c * (size_t)HW),
            (l_int*)&sA[0][r * SAP + c], 0, 0);
    }
    for (int kc0 = 0; kc0 < C; kc0 += KC) {
        const bool has_next = (kc0 + KC) < C;
        if (has_next) {
            // Prefetch next chunk into the other buffer while this one computes.
#pragma unroll
            for (int e = 0; e < EPT; ++e) {
                const int c = (e << 3) + cofs;
                __builtin_amdgcn_global_load_async_to_lds_b32(
                    (g_int*)(rowp + (size_t)(kc0 + KC + c) * (size_t)HW),
                    (l_int*)&sA[bb ^ 1][r * SAP + c], 0, 0);
            }
            __builtin_amdgcn_s_wait_asynccnt(EPT);   // chunk kc0 landed in LDS
        } else {
            __builtin_amdgcn_s_wait_asynccnt(0);
        }
        __syncthreads();                             // all waves' data visible

        const float* sAb = sA[bb];
        for (int k = 0; k < KC; k += 4) {
            v2f a;                                   // A 16x4 frag (K = 2h, 2h+1)
            const float* ap = &sAb[arow0 + k];
            a.x = ap[0];
            a.y = ap[1];
            const float* w1k = w1 + (size_t)(kc0 + k + (h << 1)) * NC;
#pragma unroll
            for (int t = 0; t < 4; ++t) {            // B 4x16 frag, col n
                const int n = nbase + (t << 4);
                v2f bf;
                bf.x = w1k[n];
                bf.y = w1k[n + NC];
                acc[t] = __builtin_amdgcn_wmma_f32_16x16x4_f32(
                    false, a, false, bf, (short)0, acc[t], false, false);
            }
        }
        __syncthreads();                             // free buffer for prefetch
        bb ^= 1;
    }
#else
    // Fallback: register-staged pipeline (next chunk's loads overlap WMMA).
    float regs[EPT];
#pragma unroll
    for (int e = 0; e < EPT; ++e) {
        const int c = (e << 3) + cofs;
        regs[e] = rowp[(size_t)c * (size_t)HW];
    }
    for (int kc0 = 0; kc0 < C; kc0 += KC) {
        __syncthreads();                             // sA free for overwrite
#pragma unroll
        for (int e = 0; e < EPT; ++e) {
            const int c = (e << 3) + cofs;
            sA[0][r * SAP + c] = regs[e];
        }
        __syncthreads();
        if ((kc0 + KC) < C) {
#pragma unroll
            for (int e = 0; e < EPT; ++e) {
                const int c = (e << 3) + cofs;
                regs[e] = rowp[(size_t)(kc0 + KC + c) * (size_t)HW];
            }
        }
        const float* sAb = sA[0];
        for (int k = 0; k < KC; k += 4) {
            v2f a;
            const float* ap = &sAb[arow0 + k];
            a.x = ap[0];
            a.y = ap[1];
            const float* w1k = w1 + (size_t)(kc0 + k + (h << 1)) * NC;
#pragma unroll
            for (int t = 0; t < 4; ++t) {
                const int n = nbase + (t << 4);
                v2f bf;
                bf.x = w1k[n];
                bf.y = w1k[n + NC];
                acc[t] = __builtin_amdgcn_wmma_f32_16x16x4_f32(
                    false, a, false, bf, (short)0, acc[t], false, false);
            }
        }
    }
#endif

    // ---------------- bias1 + ReLU -> LDS hidden ----------------
#pragma unroll
    for (int t = 0; t < 4; ++t) {
        const int n = nbase + (t << 4);
        const float bias = b1[n];
#pragma unroll
        for (int v = 0; v < 8; ++v) {          // C/D layout: row = 16*rt + 8*h + v
            const int m = (rt << 4) + (h << 3) + v;
            sH[m * SHP + n] = fmaxf(acc[t][v] + bias, 0.0f);
        }
        acc[t] = vzero;
    }
    __syncthreads();

    // ---------------- GEMM2: hidden @ w2 ----------------
    const int hrow0 = ((rt << 4) + l16) * SHP + (h << 1);
    for (int k = 0; k < NC; k += 4) {
        v2f a;
        const float* ap = &sH[hrow0 + k];
        a.x = ap[0];
        a.y = ap[1];
        const float* w2k = w2 + (size_t)(k + (h << 1)) * NC;
#pragma unroll
        for (int t = 0; t < 4; ++t) {
            const int n = nbase + (t << 4);
            v2f bf;
            bf.x = w2k[n];
            bf.y = w2k[n + NC];
            acc[t] = __builtin_amdgcn_wmma_f32_16x16x4_f32(
                false, a, false, bf, (short)0, acc[t], false, false);
        }
    }
    __syncthreads();                           // all waves done reading hidden

    // ---------------- bias2 -> LDS (overwrite hidden with pre-norm x) -------
#pragma unroll
    for (int t = 0; t < 4; ++t) {
        const int n = nbase + (t << 4);
        const float bias = b2[n];
#pragma unroll
        for (int v = 0; v < 8; ++v) {
            const int m = (rt << 4) + (h << 3) + v;
            sH[m * SHP + n] = acc[t][v] + bias;
        }
    }
    __syncthreads();

    // ---------------- row L2 norms ----------------
    {
        const int rr  = tid >> 3;              // row 0..31
        const int seg = tid & 7;               // 32-col segment
        const float* hp = &sH[rr * SHP + (seg << 5)];
        float ssum = 0.f;
#pragma unroll
        for (int j = 0; j < 32; ++j) { const float vv = hp[j]; ssum += vv * vv; }
        sRed[rr * 8 + seg] = ssum;
    }
    __syncthreads();
    if (tid < MT) {
        float ssum = 0.f;
#pragma unroll
        for (int j = 0; j < 8; ++j) ssum += sRed[tid * 8 + j];
        sRnorm[tid] = 1.0f / (sqrtf(ssum) + 1e-7f);
    }
    __syncthreads();

    // ---------------- coalesced output stores ----------------
    float* outp = out + (size_t)m0 * NC;
#pragma unroll 4
    for (int e = 0; e < MT; ++e) {
        outp[(size_t)e * NC + tid] = sH[e * SHP + tid] * sRnorm[e];
    }
}

extern "C" void kernel_launch(void* const* d_in, const int* in_sizes, int n_in,
                              void* d_out, int out_size, void* d_ws, size_t ws_size,
                              hipStream_t stream) {
    (void)in_sizes; (void)n_in; (void)out_size; (void)d_ws; (void)ws_size;

    static const int Cs[3]  = {256, 512, 1024};
    static const int HWs[3] = {128 * 128, 64 * 64, 32 * 32};

    Params p;
    for (int i = 0; i < 3; ++i) {
        p.feat[i] = (const float*)d_in[6 * i + 0];
        p.pid[i]  = (const int*)  d_in[6 * i + 1];   // JAX x64 disabled -> int32
        p.w1[i]   = (const float*)d_in[6 * i + 2];
        p.b1[i]   = (const float*)d_in[6 * i + 3];
        p.w2[i]   = (const float*)d_in[6 * i + 4];
        p.b2[i]   = (const float*)d_in[6 * i + 5];
        p.C[i]  = Cs[i];
        p.HW[i] = HWs[i];
    }
    p.out = (float*)d_out;

    dim3 grid(4096 / MT, 3);                   // 128 stripes x 3 scales
    psm_fused_kernel<<<grid, 256, 0, stream>>>(p);
}